// Network_58222576664914
// MI455X (gfx1250) — compile-verified
//
#include <hip/hip_runtime.h>

// ---- geometry (ProtoDUNE-HD-like, fixed by the reference) ----
#define TT      256
#define NCH     2560
#define NCELLS  40000
#define XSZ     (NCH * TT)        // 655360 elements per feature plane

typedef float v2f __attribute__((ext_vector_type(2)));
typedef float v8f __attribute__((ext_vector_type(8)));

// ---------------------------------------------------------------------------
// Kernel 1: relu(x) -> out plane 0, zero out planes 1..2, and fold the
// 3->8->2 affine MLP (no inner nonlinearity!) into Weff(3x2), beff(2):
//   Weff[p][o] = sum_h W1[p,h] * W2[h,o] ; beff[o] = sum_h b1[h]*W2[h,o] + b2[o]
// wcoef layout: [We00, We01, We10, We11, We20, We21, be0, be1]
// ---------------------------------------------------------------------------
__global__ void pd_prep_kernel(const float* __restrict__ x,
                               const float* __restrict__ W1,
                               const float* __restrict__ b1,
                               const float* __restrict__ W2,
                               const float* __restrict__ b2,
                               float* __restrict__ out,
                               float* __restrict__ wcoef) {
    const int i = blockIdx.x * blockDim.x + threadIdx.x;
    if (i < XSZ) {
        float r = x[i];
        r = r > 0.0f ? r : 0.0f;
        out[i]           = r;     // feature 0 = relu(x)
        out[XSZ + i]     = 0.0f;  // scattered plane 0 (amax baseline)
        out[2 * XSZ + i] = 0.0f;  // scattered plane 1
    }
    if (blockIdx.x == 0 && threadIdx.x < 8) {
        const int tid = threadIdx.x;
        float acc;
        if (tid < 6) {                 // Weff[p][o], p = tid/2, o = tid&1
            const int p = tid >> 1, o = tid & 1;
            acc = 0.0f;
            #pragma unroll
            for (int h = 0; h < 8; ++h) acc += W1[p * 8 + h] * W2[h * 2 + o];
        } else {                       // beff[o]
            const int o = tid - 6;
            acc = b2[o];
            #pragma unroll
            for (int h = 0; h < 8; ++h) acc += b1[h] * W2[h * 2 + o];
        }
        wcoef[tid] = acc;
    }
}

// ---------------------------------------------------------------------------
// Kernel 2: one block per crossing cell. Each wave computes y[o, tick] for
// 16 ticks per V_WMMA_F32_16X16X4_F32:
//   A (16x4): row0 = [We00, We10, We20, be0], row1 = [We01, We11, We21, be1]
//   B (4x16): col n = [c0(t), c1(t), c2(t), 1]
//   D (16x16): D[0,n] = y0(tick), D[1,n] = y1(tick)  -> lanes 0..15, VGPR 0/1
// B assembly is branchless: one v_cndmask selects the ch0/ch2 row offset for
// element 0 (single full-wave coalesced load), element 1 loads the ch1 row
// for all lanes (upper half hits the same cacheline) then cndmasks in 1.0.
// Scatter with amax: all surviving values are > 0 and dest starts at 0.0f,
// so signed-int atomicMax on the float bit pattern == IEEE float max.
// ---------------------------------------------------------------------------
__global__ void __launch_bounds__(256)
pd_cell_wmma_scatter(const float* __restrict__ xr,   // relu(x), plane 0 of out
                     const int*   __restrict__ gi,   // (NCELLS,3) wire idx
                     const int*   __restrict__ wc0,  // (n,2) wire->chan, plane U
                     const int*   __restrict__ wc1,  // plane V
                     const int*   __restrict__ wc2,  // plane W
                     const float* __restrict__ wcoef,
                     int* __restrict__ out1,         // int view of scattered o=0
                     int* __restrict__ out2) {       // int view of scattered o=1
    const int n        = blockIdx.x;
    const int lane     = threadIdx.x & 31;
    const int sub      = lane & 15;
    const bool lowHalf = lane < 16;
    const int waveBase = (threadIdx.x >> 5) * 32;   // 8 waves * 32 ticks

    // Per-cell channel indices (uniform per block -> scalarized loads).
    const int w0  = gi[3 * n + 0];
    const int w1  = gi[3 * n + 1];
    const int w2  = gi[3 * n + 2];
    const int ch0 = wc0[2 * w0 + 1];
    const int ch1 = wc1[2 * w1 + 1];
    const int ch2 = wc2[2 * w2 + 1];

    // Folded affine coefficients (uniform).
    const float We00 = wcoef[0], We01 = wcoef[1];
    const float We10 = wcoef[2], We11 = wcoef[3];
    const float We20 = wcoef[4], We21 = wcoef[5];
    const float be0  = wcoef[6], be1  = wcoef[7];

    // A-matrix 16x4 f32 layout (ISA 7.12.2): lanes 0-15 hold K=0 (v0), K=1 (v1);
    // lanes 16-31 hold K=2 (v0), K=3 (v1); M = lane % 16. Rows 2..15 zero.
    v2f A = {0.0f, 0.0f};
    if      (lane == 0)  { A.x = We00; A.y = We10; }   // A[0,0], A[0,1]
    else if (lane == 1)  { A.x = We01; A.y = We11; }   // A[1,0], A[1,1]
    else if (lane == 16) { A.x = We20; A.y = be0;  }   // A[0,2], A[0,3]
    else if (lane == 17) { A.x = We21; A.y = be1;  }   // A[1,2], A[1,3]

    // Branchless per-lane row offsets into xr.
    const int offA = (lowHalf ? ch0 : ch2) * TT;   // element 0: c0 / c2 row
    const int offB = ch1 * TT;                     // element 1: c1 row

    #pragma unroll
    for (int it = 0; it < 2; ++it) {
        const int base = waveBase + it * 16;

        // B-matrix 4x16 f32: lanes 0-15 -> rows K=0 (v0), K=1 (v1) at col=lane;
        // lanes 16-31 -> rows K=2 (v0), K=3 (v1) at col=lane-16.
        v2f B;
        B.x = xr[offA + base + sub];                     // full-wave load
        const float c1v = xr[offB + base + sub];         // full-wave load
        B.y = lowHalf ? c1v : 1.0f;                      // v_cndmask

        v8f C = {};
        v8f D = __builtin_amdgcn_wmma_f32_16x16x4_f32(
            /*neg_a=*/false, A, /*neg_b=*/false, B,
            /*c_mod=*/(short)0, C, /*reuse_a=*/false, /*reuse_b=*/false);

        // D VGPR0 lane n (n<16) = D[0,n] = y0(tick base+n); VGPR1 = y1.
        if (lowHalf) {
            const int t  = base + lane;
            const float y0 = D[0];
            const float y1 = D[1];
            if (y0 > 0.0f) {
                const int v = __float_as_int(y0);
                atomicMax(out1 + ch0 * TT + t, v);
                atomicMax(out1 + ch1 * TT + t, v);
                atomicMax(out1 + ch2 * TT + t, v);
            }
            if (y1 > 0.0f) {
                const int v = __float_as_int(y1);
                atomicMax(out2 + ch0 * TT + t, v);
                atomicMax(out2 + ch1 * TT + t, v);
                atomicMax(out2 + ch2 * TT + t, v);
            }
        }
    }
}

// ---------------------------------------------------------------------------
// setup_inputs() order:
//  0:x  1:W1  2:b1  3:W2  4:b2  5:gi0  6:gi1
//  7:wc00  8:wc01  9:wc02  10:wc10  11:wc11  12:wc12
// ---------------------------------------------------------------------------
extern "C" void kernel_launch(void* const* d_in, const int* in_sizes, int n_in,
                              void* d_out, int out_size, void* d_ws, size_t ws_size,
                              hipStream_t stream) {
    (void)in_sizes; (void)n_in; (void)out_size; (void)ws_size;

    const float* x  = (const float*)d_in[0];
    const float* W1 = (const float*)d_in[1];
    const float* b1 = (const float*)d_in[2];
    const float* W2 = (const float*)d_in[3];
    const float* b2 = (const float*)d_in[4];
    const int* gi[2]    = { (const int*)d_in[5], (const int*)d_in[6] };
    const int* wc[2][3] = { { (const int*)d_in[7],  (const int*)d_in[8],  (const int*)d_in[9]  },
                            { (const int*)d_in[10], (const int*)d_in[11], (const int*)d_in[12] } };

    float* out   = (float*)d_out;
    float* wcoef = (float*)d_ws;        // 8 floats of scratch

    // Phase 1: relu + zero-init scattered planes + fold MLP coefficients.
    pd_prep_kernel<<<(XSZ + 255) / 256, 256, 0, stream>>>(x, W1, b1, W2, b2, out, wcoef);

    // Phase 2: per-face cell MLP (WMMA) + amax scatter.
    int* out1 = (int*)(out + XSZ);
    int* out2 = (int*)(out + 2 * XSZ);
    for (int f = 0; f < 2; ++f) {
        pd_cell_wmma_scatter<<<NCELLS, 256, 0, stream>>>(
            out /* plane 0 = relu(x) */, gi[f],
            wc[f][0], wc[f][1], wc[f][2], wcoef, out1, out2);
    }
}